// GDN_30021821399876
// MI455X (gfx1250) — compile-verified
//
#include <hip/hip_runtime.h>
#include <math.h>

#define B_ 32
#define L_ 512
#define N_ 1024
#define D_ 64
#define TOPK_ 20
#define NODES (B_*N_)          // 32768
#define NEG_SLOPE_ 0.2f
#define BN_EPS_ 1e-5f

typedef float v2f __attribute__((ext_vector_type(2)));
typedef float v8f __attribute__((ext_vector_type(8)));

__device__ __forceinline__ float wave_sum(float v){
#pragma unroll
  for (int o = 16; o > 0; o >>= 1) v += __shfl_xor(v, o, 32);
  return v;
}
__device__ __forceinline__ float wave_max(float v){
#pragma unroll
  for (int o = 16; o > 0; o >>= 1) v = fmaxf(v, __shfl_xor(v, o, 32));
  return v;
}

// ---------------------------------------------------------------------------
// K1: per-(b,n) mean/rstd over L (no eps), write gt = normalized last step.
// ---------------------------------------------------------------------------
__global__ void k_norm_stats(const float* __restrict__ x, float* __restrict__ meanW,
                             float* __restrict__ rstdW, float* __restrict__ gt_out){
  int n = blockIdx.x * 256 + threadIdx.x;   // 0..1023
  int b = blockIdx.y;                       // 0..31
  const float* p = x + (size_t)b * L_ * N_ + n;
  float s = 0.f, s2 = 0.f, last = 0.f;
  for (int l = 0; l < L_; ++l){
    float v = p[(size_t)l * N_];
    s += v; s2 += v * v;
    if (l == L_ - 1) last = v;
  }
  float m   = s  * (1.f / L_);
  float var = s2 * (1.f / L_) - m * m;
  float rs  = rsqrtf(var);                  // no eps (reference has none here)
  meanW[b * N_ + n] = m;
  rstdW[b * N_ + n] = rs;
  float g = (last - m) * rs;
  if (!__builtin_isfinite(g)) g = 0.f;      // nan_to_num
  gt_out[b * N_ + n] = g;
}

// ---------------------------------------------------------------------------
// K2a: row norms of emb. One wave per row.
// ---------------------------------------------------------------------------
__global__ void k_norms(const float* __restrict__ emb, float* __restrict__ nrm){
  int lane = threadIdx.x & 31, wid = threadIdx.x >> 5;
  int j = blockIdx.x * 8 + wid;
  float e0 = emb[j * D_ + lane];
  float e1 = emb[j * D_ + lane + 32];
  float s = wave_sum(e0 * e0 + e1 * e1);
  if (lane == 0) nrm[j] = sqrtf(s);
}

// ---------------------------------------------------------------------------
// K2b: top-20 of cosine similarity per row. One block (256 thr) per row.
// Iterative argmax extraction, lower-index tie-break (matches lax.top_k).
// ---------------------------------------------------------------------------
__global__ void k_topk(const float* __restrict__ emb, const float* __restrict__ nrm,
                       int* __restrict__ topk){
  __shared__ float rowi[D_];
  __shared__ float sval[256];
  __shared__ int   sidx[256];
  int i = blockIdx.x, tid = threadIdx.x;
  if (tid < D_) rowi[tid] = emb[i * D_ + tid];
  __syncthreads();
  float invni = 1.f / nrm[i];
  float vals[4];
#pragma unroll
  for (int q = 0; q < 4; ++q){
    int j = tid + 256 * q;
    const float* pj = emb + j * D_;
    float dot = 0.f;
    for (int d = 0; d < D_; ++d) dot += rowi[d] * pj[d];
    vals[q] = dot * invni / nrm[j];
  }
  for (int t = 0; t < TOPK_; ++t){
    float bv = vals[0]; int bj = tid;        // ascending q + strict '>' keeps lowest idx
#pragma unroll
    for (int q = 1; q < 4; ++q){
      int j = tid + 256 * q;
      if (vals[q] > bv){ bv = vals[q]; bj = j; }
    }
    sval[tid] = bv; sidx[tid] = bj;
    __syncthreads();
    for (int s = 128; s > 0; s >>= 1){
      if (tid < s){
        float v2 = sval[tid + s]; int j2 = sidx[tid + s];
        if (v2 > sval[tid] || (v2 == sval[tid] && j2 < sidx[tid])){
          sval[tid] = v2; sidx[tid] = j2;
        }
      }
      __syncthreads();
    }
    int win = sidx[0];
    if (tid == 0) topk[i * TOPK_ + t] = win;
    if ((win & 255) == tid) vals[win >> 8] = -INFINITY;
    __syncthreads();
  }
}

// ---------------------------------------------------------------------------
// K3: h = X @ lin_W via V_WMMA_F32_16X16X4_F32.
// X row i = normalized enc[(i%64)/2, i/64, ((i%64)&1)*512 : +512], with row
// l==511 zeroed (masked timestep). Normalization + nan_to_num fused into the
// LDS staging, avoiding a 64MB materialized intermediate.
// Block = 256 thr (8 waves): 128 rows x 64 cols; K chunked by 64 via LDS.
// LDS strides 137 / 73 -> half-wave bank sets disjoint mod 64 on reads,
// all-64-bank cooperative writes.
// ---------------------------------------------------------------------------
#define AS_ 137
#define WS_ 73
__global__ void k_gemm_h(const float* __restrict__ x, const float* __restrict__ meanW,
                         const float* __restrict__ rstdW, const float* __restrict__ linW,
                         float* __restrict__ h){
  __shared__ float At[64 * AS_];   // [k][m], m = 0..127
  __shared__ float Wl[64 * WS_];   // [k][d], d = 0..63
  int tid  = threadIdx.x;
  int lane = tid & 31, wid = tid >> 5;
  int row0 = blockIdx.x * 128;

  v8f zero = {0.f,0.f,0.f,0.f,0.f,0.f,0.f,0.f};
  v8f acc[4] = {zero, zero, zero, zero};

  for (int k0 = 0; k0 < 512; k0 += 64){
    for (int idx = tid; idx < 64 * 64; idx += 256){
      int r = idx >> 6, c = idx & 63;
      Wl[r * WS_ + c] = linW[(k0 + r) * 64 + c];
    }
    for (int idx = tid; idx < 128 * 64; idx += 256){
      int row = idx >> 6, kk = idx & 63;
      int i  = row0 + row;
      int lp = i >> 6;                 // timestep l
      int rr = i & 63;
      int bp = rr >> 1;                // batch
      int n  = ((rr & 1) << 9) + k0 + kk;
      float a;
      if (lp == L_ - 1){
        a = 0.f;                       // enc[:, -1, :] zeroed
      } else {
        float v = x[((size_t)(bp * L_ + lp)) * N_ + n];
        a = (v - meanW[bp * N_ + n]) * rstdW[bp * N_ + n];
        if (!__builtin_isfinite(a)) a = 0.f;   // nan_to_num
      }
      At[kk * AS_ + row] = a;
    }
    __syncthreads();

    int m    = wid * 16 + (lane & 15);
    int koff = (lane >> 4) << 1;       // 0 or 2 per ISA 16x4 f32 A layout
#pragma unroll
    for (int kk = 0; kk < 64; kk += 4){
      v2f a;
      a.x = At[(kk + koff    ) * AS_ + m];
      a.y = At[(kk + koff + 1) * AS_ + m];
#pragma unroll
      for (int ct = 0; ct < 4; ++ct){
        v2f bb;
        int c = ct * 16 + (lane & 15);
        bb.x = Wl[(kk + koff    ) * WS_ + c];
        bb.y = Wl[(kk + koff + 1) * WS_ + c];
        acc[ct] = __builtin_amdgcn_wmma_f32_16x16x4_f32(
            false, a, false, bb, (short)0, acc[ct], false, false);
      }
    }
    __syncthreads();
  }

  int halfm = (lane >> 4) * 8;
  int cc    = lane & 15;
#pragma unroll
  for (int ct = 0; ct < 4; ++ct){
#pragma unroll
    for (int r = 0; r < 8; ++r){
      int mrow = row0 + wid * 16 + halfm + r;
      h[(size_t)mrow * 64 + ct * 16 + cc] = acc[ct][r];
    }
  }
}

// ---------------------------------------------------------------------------
// K4: per-node attention scalars a_i, c_j. One wave per node.
// ---------------------------------------------------------------------------
__global__ void k_node_scalars(const float* __restrict__ h, const float* __restrict__ emb,
    const float* __restrict__ atti, const float* __restrict__ attj,
    const float* __restrict__ attemi, const float* __restrict__ attemj,
    float* __restrict__ ai, float* __restrict__ cj){
  int lane = threadIdx.x & 31, wid = threadIdx.x >> 5;
  int node = blockIdx.x * 8 + wid;
  int n = node & (N_ - 1);
  float h0 = h[(size_t)node * 64 + lane];
  float h1 = h[(size_t)node * 64 + lane + 32];
  float e0 = emb[n * 64 + lane];
  float e1 = emb[n * 64 + lane + 32];
  float va = h0 * atti[lane] + h1 * atti[lane + 32] + e0 * attemi[lane] + e1 * attemi[lane + 32];
  float vc = h0 * attj[lane] + h1 * attj[lane + 32] + e0 * attemj[lane] + e1 * attemj[lane + 32];
  va = wave_sum(va);
  vc = wave_sum(vc);
  if (lane == 0){ ai[node] = va; cj[node] = vc; }
}

// ---------------------------------------------------------------------------
// K5: leaky-relu logits, masked softmax over 20 topk edges + self-loop,
// weighted aggregation of h, + gnn_b. One wave per destination node.
// h (8MB) is L2-resident on MI455X (192MB L2) so gathers are cheap.
// ---------------------------------------------------------------------------
__global__ void k_attn_aggregate(const float* __restrict__ h, const int* __restrict__ topk,
    const float* __restrict__ ai, const float* __restrict__ cj,
    const float* __restrict__ gnnb, float* __restrict__ outw){
  int lane = threadIdx.x & 31, wid = threadIdx.x >> 5;
  int dst = blockIdx.x * 8 + wid;
  int b = dst >> 10, n = dst & (N_ - 1);
  float adst = ai[dst];
  float logit = -INFINITY;
  int   srcn  = dst;
  bool  valid = false;
  if (lane < TOPK_){
    int s = topk[n * TOPK_ + lane];
    if (s != n){                        // self_mask -> -inf (skip)
      srcn = b * N_ + s;
      float lg = adst + cj[srcn];
      logit = (lg >= 0.f) ? lg : NEG_SLOPE_ * lg;
      valid = true;
    }
  } else if (lane == TOPK_){            // appended self-loop edge (unmasked)
    float lg = adst + cj[dst];
    logit = (lg >= 0.f) ? lg : NEG_SLOPE_ * lg;
    valid = true;
  }
  float mx    = wave_max(logit);
  float e     = valid ? expf(logit - mx) : 0.f;
  float denom = wave_sum(e);
  float alpha = e / denom;
  float acc0 = 0.f, acc1 = 0.f;
  for (int k = 0; k <= TOPK_; ++k){
    float ak = __shfl(alpha, k, 32);
    int   sk = __shfl(srcn,  k, 32);
    if (ak != 0.f){
      acc0 += ak * h[(size_t)sk * 64 + lane];
      acc1 += ak * h[(size_t)sk * 64 + lane + 32];
    }
  }
  outw[(size_t)dst * 64 + lane]      = acc0 + gnnb[lane];
  outw[(size_t)dst * 64 + lane + 32] = acc1 + gnnb[lane + 32];
}

// ---------------------------------------------------------------------------
// K6: per-feature mean / rstd over all 32768 rows. One block per feature.
// ---------------------------------------------------------------------------
__global__ void k_feat_stats(const float* __restrict__ src, float* __restrict__ stats, float eps){
  __shared__ float ss[256], ss2[256];
  int d = blockIdx.x, tid = threadIdx.x;
  float s = 0.f, s2 = 0.f;
  for (int r = tid; r < NODES; r += 256){
    float v = src[(size_t)r * 64 + d];
    s += v; s2 += v * v;
  }
  ss[tid] = s; ss2[tid] = s2;
  __syncthreads();
  for (int t = 128; t > 0; t >>= 1){
    if (tid < t){ ss[tid] += ss[tid + t]; ss2[tid] += ss2[tid + t]; }
    __syncthreads();
  }
  if (tid == 0){
    float m   = ss[0]  * (1.f / NODES);
    float var = ss2[0] * (1.f / NODES) - m * m;
    stats[d]      = m;
    stats[64 + d] = rsqrtf(var + eps);
  }
}

// ---------------------------------------------------------------------------
// K7: y = relu(BN1(out)) * emb[n]   (elementwise, coalesced)
// ---------------------------------------------------------------------------
__global__ void k_bn1_mul(const float* __restrict__ outw, const float* __restrict__ stats1,
    const float* __restrict__ g1, const float* __restrict__ b1,
    const float* __restrict__ emb, float* __restrict__ y){
  int gid  = blockIdx.x * 256 + threadIdx.x;
  int d    = gid & 63;
  int node = gid >> 6;
  int n    = node & (N_ - 1);
  float v = outw[gid];
  v = (v - stats1[d]) * stats1[64 + d] * g1[d] + b1[d];
  v = fmaxf(v, 0.f);
  y[gid] = v * emb[n * 64 + d];
}

// ---------------------------------------------------------------------------
// K8: o[node] = sum_d relu(BN2(y)) * out_W[d] + out_b. One wave per node.
// ---------------------------------------------------------------------------
__global__ void k_final(const float* __restrict__ y, const float* __restrict__ stats2,
    const float* __restrict__ g2, const float* __restrict__ b2,
    const float* __restrict__ outW, const float* __restrict__ outb,
    float* __restrict__ out){
  int lane = threadIdx.x & 31, wid = threadIdx.x >> 5;
  int node = blockIdx.x * 8 + wid;
  float acc = 0.f;
#pragma unroll
  for (int q = 0; q < 2; ++q){
    int d = lane + q * 32;
    float v = y[(size_t)node * 64 + d];
    v = (v - stats2[d]) * stats2[64 + d] * g2[d] + b2[d];
    v = fmaxf(v, 0.f);
    acc += v * outW[d];
  }
  acc = wave_sum(acc);
  if (lane == 0) out[node] = acc + outb[0];
}

// ---------------------------------------------------------------------------
extern "C" void kernel_launch(void* const* d_in, const int* in_sizes, int n_in,
                              void* d_out, int out_size, void* d_ws, size_t ws_size,
                              hipStream_t stream){
  (void)in_sizes; (void)n_in; (void)out_size; (void)ws_size;
  const float* x_enc  = (const float*)d_in[0];
  const float* emb    = (const float*)d_in[1];
  const float* linW   = (const float*)d_in[2];
  const float* atti   = (const float*)d_in[3];
  const float* attj   = (const float*)d_in[4];
  const float* attemi = (const float*)d_in[5];
  const float* attemj = (const float*)d_in[6];
  const float* gnnb   = (const float*)d_in[7];
  const float* g1     = (const float*)d_in[8];
  const float* b1     = (const float*)d_in[9];
  const float* g2     = (const float*)d_in[10];
  const float* b2     = (const float*)d_in[11];
  const float* outW   = (const float*)d_in[12];
  const float* outb   = (const float*)d_in[13];
  float* out = (float*)d_out;          // [0,32768): o ; [32768,65536): gt

  float* W      = (float*)d_ws;        // ~25.8 MB total scratch
  float* meanW  = W;                   // 32768
  float* rstdW  = W + 32768;           // 32768
  float* nrm    = W + 65536;           // 1024
  int*   topk   = (int*)(W + 66560);   // 20480 ints
  float* h      = W + 87040;           // 2097152
  float* ai     = W + 2184192;         // 32768
  float* cj     = W + 2216960;         // 32768
  float* outw   = W + 2249728;         // 2097152
  float* y      = W + 4346880;         // 2097152
  float* stats1 = W + 6444032;         // 128
  float* stats2 = W + 6444160;         // 128

  k_norm_stats    <<<dim3(4, 32), 256, 0, stream>>>(x_enc, meanW, rstdW, out + NODES);
  k_norms         <<<128,  256, 0, stream>>>(emb, nrm);
  k_topk          <<<1024, 256, 0, stream>>>(emb, nrm, topk);
  k_gemm_h        <<<256,  256, 0, stream>>>(x_enc, meanW, rstdW, linW, h);
  k_node_scalars  <<<4096, 256, 0, stream>>>(h, emb, atti, attj, attemi, attemj, ai, cj);
  k_attn_aggregate<<<4096, 256, 0, stream>>>(h, topk, ai, cj, gnnb, outw);
  k_feat_stats    <<<64,   256, 0, stream>>>(outw, stats1, BN_EPS_);
  k_bn1_mul       <<<8192, 256, 0, stream>>>(outw, stats1, g1, b1, emb, y);
  k_feat_stats    <<<64,   256, 0, stream>>>(y, stats2, BN_EPS_);
  k_final         <<<4096, 256, 0, stream>>>(y, stats2, g2, b2, outW, outb, out);
}